// res_gcn_up_15616501088319
// MI455X (gfx1250) — compile-verified
//
#include <hip/hip_runtime.h>
#include <hip/hip_bf16.h>

// Problem constants (match reference)
#define B_  4
#define N_  16384
#define C_  128
#define K_  8
#define BN_ (B_ * N_)
#define UP_ 2

typedef __attribute__((ext_vector_type(2))) float v2f;
typedef __attribute__((ext_vector_type(8))) float v8f;

// ---------------------------------------------------------------------------
// 1) Transpose points [B,C,N] -> P [B,N,C] (LDS-tiled, coalesced both sides)
// ---------------------------------------------------------------------------
__global__ void __launch_bounds__(256)
transpose_cn_nc(const float* __restrict__ pts, float* __restrict__ P) {
    __shared__ float tile[32][33];
    const int b  = blockIdx.z;
    const int n0 = blockIdx.x * 32;
    const int c0 = blockIdx.y * 32;
    const int tx = threadIdx.x;       // 0..31
    const int ty = threadIdx.y;       // 0..7
#pragma unroll
    for (int j = 0; j < 4; ++j) {
        tile[ty + 8 * j][tx] = pts[((size_t)b * C_ + (c0 + ty + 8 * j)) * N_ + n0 + tx];
    }
    __syncthreads();
#pragma unroll
    for (int j = 0; j < 4; ++j) {
        P[((size_t)b * N_ + (n0 + ty + 8 * j)) * C_ + c0 + tx] = tile[tx][ty + 8 * j];
    }
}

// ---------------------------------------------------------------------------
// 2) Gather-sum with fused ReLU:
//    S[g, c] = sum_k relu(P[b, idx[g,k], c])   (one wave per point; each
//    neighbor row is a contiguous 512B read -> 32 lanes x float4, coalesced)
// ---------------------------------------------------------------------------
__global__ void __launch_bounds__(256)
gather_sum_relu(const float* __restrict__ P, const int* __restrict__ idx,
                float* __restrict__ S) {
    const int wave = threadIdx.x >> 5;
    const int lane = threadIdx.x & 31;
    const int g    = blockIdx.x * 8 + wave;      // global point id in [0, BN)
    const int b    = g >> 14;                    // / N_
    const int c4   = lane * 4;

    const int* ip = idx + (size_t)g * K_;
    const int4 i0 = *(const int4*)(ip);
    const int4 i1 = *(const int4*)(ip + 4);
    const int nb[8] = {i0.x, i0.y, i0.z, i0.w, i1.x, i1.y, i1.z, i1.w};

    const float* Pb = P + (size_t)b * N_ * C_;
    float4 acc = make_float4(0.f, 0.f, 0.f, 0.f);
#pragma unroll
    for (int k = 0; k < K_; ++k) {
        const float4 v = *(const float4*)(Pb + (size_t)nb[k] * C_ + c4);
        acc.x += fmaxf(v.x, 0.f);
        acc.y += fmaxf(v.y, 0.f);
        acc.z += fmaxf(v.z, 0.f);
        acc.w += fmaxf(v.w, 0.f);
    }
    *(float4*)(S + (size_t)g * C_ + c4) = acc;
}

// ---------------------------------------------------------------------------
// 3) Residual-GCN GEMM via V_WMMA_F32_16X16X4_F32:
//    Pout[p, o] = (sum_c relu(P[p,c])*W1[o,c] + S[p,c]*W2[o,c]) / 9 + P[p,o]
//    Block: 32 points x 128 outs; 8 waves, each wave owns a 16-out column and
//    carries TWO independent accumulator tiles (points 0-15 / 16-31) sharing
//    one B fragment -> two interleaved WMMA chains, latency hidden.
//    X tile staged global->LDS with gfx1250 async copies (ASYNCcnt-tracked);
//    ReLU applied at A-fragment read for the feats chunks.
// ---------------------------------------------------------------------------
__global__ void __launch_bounds__(256)
gcn_gemm_wmma(const float* __restrict__ P, const float* __restrict__ S,
              const float* __restrict__ W1, const float* __restrict__ W2,
              float* __restrict__ Pout) {
    __shared__ float Xs[32][68];                 // 32 pts x 64 ch (+4 pad: 16B rows)

    const int pb   = blockIdx.x * 32;            // first point of this block
    const int wave = threadIdx.x >> 5;
    const int lane = threadIdx.x & 31;
    const int o0   = wave * 16;                  // out-channel base for wave
    const int half = lane >> 4;                  // A/B: K-half select (ISA layout)
    const int m    = lane & 15;                  // A row / B column within tile

    // staging coords: 256 threads move 32x64 floats (2 x float4 each)
    const int sp = threadIdx.x >> 3;             // point row 0..31
    const int sc = (threadIdx.x & 7) * 8;        // channel octet 0..56

    v8f acc0 = {};
    v8f acc1 = {};

#pragma unroll
    for (int chunk = 0; chunk < 4; ++chunk) {    // K = 256 in chunks of 64
        const int  cbase  = chunk * 64;
        const bool isFeat = (cbase < 128);       // feats region needs ReLU
        {
            const size_t rb  = (size_t)(pb + sp) * C_;
            const float* src = isFeat ? (P + rb + cbase + sc)
                                      : (S + rb + (cbase - 128) + sc);
            const uint32_t l0 = (uint32_t)(uintptr_t)(&Xs[sp][sc]);
            const uint32_t l1 = l0 + 16;
            const float*   s1 = src + 4;
            // gfx1250 async global->LDS copies (tracked with ASYNCcnt)
            asm volatile("global_load_async_to_lds_b128 %0, %1, off"
                         :: "v"(l0), "v"(src) : "memory");
            asm volatile("global_load_async_to_lds_b128 %0, %1, off"
                         :: "v"(l1), "v"(s1) : "memory");
            if (chunk < 3) {                     // prefetch next chunk's weights
                const int    nc = cbase + 64;
                const float* nW = (nc < 128) ? W1 : W2;
                __builtin_prefetch(nW + (size_t)(o0 + m) * C_ + (nc & 127), 0, 0);
            }
            asm volatile("s_wait_asynccnt 0x0" ::: "memory");
        }
        __syncthreads();

        const float* Wsel  = isFeat ? W1 : W2;
        const int    crow0 = isFeat ? cbase : (cbase - 128);
        const int    o     = o0 + m;
#pragma unroll
        for (int kk = 0; kk < 16; ++kk) {        // 16 x (K=4) WMMA steps
            const int ca = kk * 4 + half * 2;    // channel within chunk
            float x0 = Xs[m][ca],      x1 = Xs[m][ca + 1];
            float y0 = Xs[m + 16][ca], y1 = Xs[m + 16][ca + 1];
            if (isFeat) {                        // ReLU shifted to A-read
                x0 = fmaxf(x0, 0.f); x1 = fmaxf(x1, 0.f);
                y0 = fmaxf(y0, 0.f); y1 = fmaxf(y1, 0.f);
            }
            v2f a0 = {x0, x1};
            v2f a1 = {y0, y1};
            const int cr = crow0 + kk * 4 + half * 2;
            v2f bb;
            bb[0] = Wsel[(size_t)o * C_ + cr];
            bb[1] = Wsel[(size_t)o * C_ + cr + 1];
            acc0 = __builtin_amdgcn_wmma_f32_16x16x4_f32(
                false, a0, false, bb, (short)0, acc0, false, false);
            acc1 = __builtin_amdgcn_wmma_f32_16x16x4_f32(
                false, a1, false, bb, (short)0, acc1, false, false);
        }
        __syncthreads();
    }

    // Epilogue: /(K+1) + shortcut. C/D layout: VGPR v -> row v (lanes 0-15)
    // or row v+8 (lanes 16-31); column = lane & 15.
    const float inv = 1.0f / (float)(K_ + 1);
#pragma unroll
    for (int v = 0; v < 8; ++v) {
        const int    row = v + half * 8;
        const size_t a0  = (size_t)(pb + row) * C_ + o0 + m;
        const size_t a1  = (size_t)(pb + 16 + row) * C_ + o0 + m;
        Pout[a0] = acc0[v] * inv + P[a0];
        Pout[a1] = acc1[v] * inv + P[a1];
    }
}

// ---------------------------------------------------------------------------
// 4) Final upsampling head: 6 output channels per point
//    delta[o] = (Wc[o,:]·relu(P[g,:]) + Wg[o,:]·S[g,:]) / 9
//    out[b, r*N + n, j] = delta[r*3+j] + xyz[b, n, j]
// ---------------------------------------------------------------------------
__global__ void __launch_bounds__(256)
final_upsample(const float* __restrict__ P, const float* __restrict__ S,
               const float* __restrict__ Wc, const float* __restrict__ Wg,
               const float* __restrict__ xyz, float* __restrict__ out) {
    const int wave = threadIdx.x >> 5;
    const int lane = threadIdx.x & 31;
    const int g    = blockIdx.x * 8 + wave;
    const int b    = g >> 14;
    const int n    = g & (N_ - 1);
    const int c4   = lane * 4;

    float4 xp = *(const float4*)(P + (size_t)g * C_ + c4);
    xp.x = fmaxf(xp.x, 0.f); xp.y = fmaxf(xp.y, 0.f);
    xp.z = fmaxf(xp.z, 0.f); xp.w = fmaxf(xp.w, 0.f);
    const float4 xs = *(const float4*)(S + (size_t)g * C_ + c4);

    float d[6];
#pragma unroll
    for (int o = 0; o < 6; ++o) {
        const float4 wc = *(const float4*)(Wc + o * C_ + c4);
        const float4 wg = *(const float4*)(Wg + o * C_ + c4);
        float p = xp.x * wc.x + xp.y * wc.y + xp.z * wc.z + xp.w * wc.w
                + xs.x * wg.x + xs.y * wg.y + xs.z * wg.z + xs.w * wg.w;
#pragma unroll
        for (int off = 16; off > 0; off >>= 1) p += __shfl_xor(p, off, 32);
        d[o] = p;   // all lanes hold the full sum after butterfly
    }
    if (lane < 6) {
        const int   o     = lane;
        const int   r     = o / 3;
        const int   j     = o - r * 3;
        const float delta = d[o] * (1.0f / (float)(K_ + 1));
        out[((size_t)b * (UP_ * N_) + (size_t)r * N_ + n) * 3 + j] =
            delta + xyz[((size_t)b * N_ + n) * 3 + j];
    }
}

// ---------------------------------------------------------------------------
// Launch: transpose -> 2x (gather_sum, wmma GEMM, ping-pong) -> gather_sum ->
// final head. Workspace: 3 x BN*C floats (~100 MB).
// ---------------------------------------------------------------------------
extern "C" void kernel_launch(void* const* d_in, const int* in_sizes, int n_in,
                              void* d_out, int out_size, void* d_ws, size_t ws_size,
                              hipStream_t stream) {
    (void)in_sizes; (void)n_in; (void)out_size; (void)ws_size;
    const float* points  = (const float*)d_in[0];
    const float* xyz     = (const float*)d_in[1];
    const int*   indices = (const int*)d_in[2];
    const float* W1      = (const float*)d_in[3];   // [2,128,128]
    const float* W2      = (const float*)d_in[4];   // [2,128,128]
    const float* Wc      = (const float*)d_in[5];   // [6,128]
    const float* Wg      = (const float*)d_in[6];   // [6,128]
    float*       out     = (float*)d_out;

    float* P0 = (float*)d_ws;
    float* P1 = P0 + (size_t)BN_ * C_;
    float* S  = P1 + (size_t)BN_ * C_;

    {
        dim3 tb(32, 8, 1);
        dim3 tg(N_ / 32, C_ / 32, B_);
        transpose_cn_nc<<<tg, tb, 0, stream>>>(points, P0);
    }

    float* Pc = P0;
    float* Pn = P1;
    for (int i = 0; i < 2; ++i) {
        gather_sum_relu<<<BN_ / 8, 256, 0, stream>>>(Pc, indices, S);
        gcn_gemm_wmma<<<BN_ / 32, 256, 0, stream>>>(
            Pc, S, W1 + (size_t)i * C_ * C_, W2 + (size_t)i * C_ * C_, Pn);
        float* t = Pc; Pc = Pn; Pn = t;
    }
    gather_sum_relu<<<BN_ / 8, 256, 0, stream>>>(Pc, indices, S);
    final_upsample<<<BN_ / 8, 256, 0, stream>>>(Pc, S, Wc, Wg, xyz, out);
}